// Client_70360154243704
// MI455X (gfx1250) — compile-verified
//
#include <hip/hip_runtime.h>

typedef __attribute__((ext_vector_type(2))) float v2f;
typedef __attribute__((ext_vector_type(4))) float v4f;
typedef __attribute__((ext_vector_type(8))) float v8f;

#define B_SZ 64
#define T_SZ 4096
#define D_DIM 100
#define N_ROWS (B_SZ * T_SZ)      // 262144
#define K_CODES 4
#define BETA 0.25f

#define THREADS 256               // 8 waves (wave32)
#define ROWS_PER_BLOCK 128        // 16 rows per wave
#define COL_TILES 7               // 7*16 = 112 >= 100 (padded)
#define DPAD 112

// LDS layout in floats
#define LDS_X_OFF  0                          // 128*100 = 12800
#define LDS_W_OFF  (ROWS_PER_BLOCK * D_DIM)   // 112*100 = 11200 (rows 100..111 zero)
#define LDS_B_OFF  (LDS_W_OFF + DPAD * D_DIM) // 112
#define LDS_CB_OFF (LDS_B_OFF + DPAD)         // 4*112 = 448 (cols 100..111 zero)
#define LDS_CN_OFF (LDS_CB_OFF + K_CODES * DPAD) // 4
#define LDS_FLOATS (LDS_CN_OFF + K_CODES)
#define LDS_BYTES  (LDS_FLOATS * 4)

__global__ void init_loss_kernel(float* loss) {
  if (threadIdx.x == 0) *loss = 0.0f;
}

__global__ void __launch_bounds__(THREADS)
vqvae_fused_kernel(const float* __restrict__ x,
                   const float* __restrict__ W1,
                   const float* __restrict__ b1,
                   const float* __restrict__ cbk,
                   const int*   __restrict__ qflag,
                   float* __restrict__ h_out,
                   float* __restrict__ payload,
                   float* __restrict__ aux,
                   float* __restrict__ loss)
{
  extern __shared__ float smem[];
  float* lds_x  = smem + LDS_X_OFF;
  float* lds_w  = smem + LDS_W_OFF;
  float* lds_b  = smem + LDS_B_OFF;
  float* lds_cb = smem + LDS_CB_OFF;
  float* lds_cn = smem + LDS_CN_OFF;

  const int tid      = threadIdx.x;
  const int blockRow = blockIdx.x * ROWS_PER_BLOCK;

  // ---------------- cooperative staging (float4-coalesced) ----------------
  {
    const v4f* xg = (const v4f*)(x + (size_t)blockRow * D_DIM);
    v4f* xl = (v4f*)lds_x;
    for (int i = tid; i < (ROWS_PER_BLOCK * D_DIM) / 4; i += THREADS)
      xl[i] = xg[i];

    const v4f* wg = (const v4f*)W1;
    v4f* wl = (v4f*)lds_w;
    for (int i = tid; i < (D_DIM * D_DIM) / 4; i += THREADS)
      wl[i] = wg[i];
    for (int i = D_DIM * D_DIM + tid; i < DPAD * D_DIM; i += THREADS)
      lds_w[i] = 0.0f;                       // zero-pad cols 100..111 of h

    if (tid < DPAD) lds_b[tid] = (tid < D_DIM) ? b1[tid] : 0.0f;

    for (int i = tid; i < K_CODES * DPAD; i += THREADS) {
      int k = i / DPAD, c = i - k * DPAD;
      lds_cb[i] = (c < D_DIM) ? cbk[k * D_DIM + c] : 0.0f;
    }
    if (tid < K_CODES) {                     // ||c_k||^2
      float s = 0.0f;
      for (int d = 0; d < D_DIM; ++d) {
        float v = cbk[tid * D_DIM + d];
        s += v * v;
      }
      lds_cn[tid] = s;
    }
  }
  __syncthreads();

  const int wave = tid >> 5;
  const int lane = tid & 31;
  const int half = lane >> 4;   // 0: lanes 0-15 (K=k0,k0+1 / rows M=r), 1: lanes 16-31
  const int l16  = lane & 15;
  const int qz   = qflag[0];

  // ------------- WMMA f32 16x16x4 GEMM: 16 rows x 112 cols per wave -------------
  v8f acc[COL_TILES] = {};
  const float* aBase = lds_x + (wave * 16 + l16) * D_DIM + half * 2;

  for (int k0 = 0; k0 < D_DIM; k0 += 4) {
    // A 16x4 layout: lane<16 holds K=k0,k0+1 of row l16; lane>=16 holds K=k0+2,k0+3
    v2f a = *(const v2f*)(aBase + k0);
    #pragma unroll
    for (int c = 0; c < COL_TILES; ++c) {
      // B 4x16 layout (B[k][e] = W1[e][k]): vgpr0 rows k0/k0+2, vgpr1 rows k0+1/k0+3
      const float* bp = lds_w + (c * 16 + l16) * D_DIM + half * 2 + k0;
      v2f b = *(const v2f*)bp;
      acc[c] = __builtin_amdgcn_wmma_f32_16x16x4_f32(
          /*neg_a=*/false, a, /*neg_b=*/false, b,
          /*c_mod=*/(short)0, acc[c], /*reuse_a=*/false, /*reuse_b=*/false);
    }
  }

  // bias + ReLU (padded cols have bias 0, W 0 -> stay 0)
  #pragma unroll
  for (int c = 0; c < COL_TILES; ++c) {
    float bias = lds_b[c * 16 + l16];
    #pragma unroll
    for (int r = 0; r < 8; ++r)
      acc[c][r] = fmaxf(acc[c][r] + bias, 0.0f);
  }

  // ------------- fused VQ: per-row argmin over 4 codes + outputs -------------
  const int   waveRow   = blockRow + wave * 16;
  const float lossScale = (1.0f + BETA) / ((float)N_ROWS * (float)D_DIM);
  float lsum = 0.0f;

  #pragma unroll
  for (int r = 0; r < 8; ++r) {
    const int gRow = waveRow + half * 8 + r;   // C/D layout: vgpr r = row r (+8 for upper half)
    float dk0 = 0.0f, dk1 = 0.0f, dk2 = 0.0f, dk3 = 0.0f;
    #pragma unroll
    for (int c = 0; c < COL_TILES; ++c) {
      const int col = c * 16 + l16;
      const float hv = acc[c][r];
      if (col < D_DIM)
        h_out[(size_t)gRow * D_DIM + col] = hv;
      dk0 += hv * lds_cb[0 * DPAD + col];      // padded cols contribute 0
      dk1 += hv * lds_cb[1 * DPAD + col];
      dk2 += hv * lds_cb[2 * DPAD + col];
      dk3 += hv * lds_cb[3 * DPAD + col];
    }
    // reduce partial dots across the 16 lanes holding this row
    #pragma unroll
    for (int m = 1; m < 16; m <<= 1) {
      dk0 += __shfl_xor(dk0, m, 16);
      dk1 += __shfl_xor(dk1, m, 16);
      dk2 += __shfl_xor(dk2, m, 16);
      dk3 += __shfl_xor(dk3, m, 16);
    }
    if (qz) {
      // argmin_k ||c_k||^2 - 2 h.c_k  (||h||^2 constant per row; first-min wins)
      int best = 0;
      float bv = lds_cn[0] - 2.0f * dk0;
      float v1 = lds_cn[1] - 2.0f * dk1; if (v1 < bv) { bv = v1; best = 1; }
      float v2 = lds_cn[2] - 2.0f * dk2; if (v2 < bv) { bv = v2; best = 2; }
      float v3 = lds_cn[3] - 2.0f * dk3; if (v3 < bv) { bv = v3; best = 3; }
      if (l16 == 0) payload[gRow] = (float)best;
      #pragma unroll
      for (int c = 0; c < COL_TILES; ++c) {
        const int col = c * 16 + l16;
        if (col < D_DIM) {
          float q  = lds_cb[best * DPAD + col]; // aux = q (straight-through forward)
          float dd = acc[c][r] - q;
          aux[(size_t)gRow * D_DIM + col] = q;
          lsum += dd * dd;
        }
      }
    } else {
      if (l16 == 0) payload[gRow] = 0.0f;
      #pragma unroll
      for (int c = 0; c < COL_TILES; ++c) {
        const int col = c * 16 + l16;
        if (col < D_DIM)
          aux[(size_t)gRow * D_DIM + col] = 0.0f;
      }
    }
  }

  // wave-level loss reduction, one atomic per wave
  #pragma unroll
  for (int m = 1; m < 32; m <<= 1)
    lsum += __shfl_xor(lsum, m, 32);
  if (lane == 0 && qz)
    atomicAdd(loss, lsum * lossScale);
}

extern "C" void kernel_launch(void* const* d_in, const int* in_sizes, int n_in,
                              void* d_out, int out_size, void* d_ws, size_t ws_size,
                              hipStream_t stream) {
  (void)in_sizes; (void)n_in; (void)out_size; (void)d_ws; (void)ws_size;
  const float* x   = (const float*)d_in[0];
  const float* W1  = (const float*)d_in[1];
  const float* b1  = (const float*)d_in[2];
  const float* cbk = (const float*)d_in[3];
  const int*   qz  = (const int*)d_in[4];

  float* out     = (float*)d_out;
  float* h       = out;                                   // B*T*D
  float* payload = h + (size_t)N_ROWS * D_DIM;            // B*T
  float* aux     = payload + N_ROWS;                      // B*T*D
  float* loss    = aux + (size_t)N_ROWS * D_DIM;          // 1

  init_loss_kernel<<<1, 32, 0, stream>>>(loss);

  const int blocks = N_ROWS / ROWS_PER_BLOCK;             // 2048
  vqvae_fused_kernel<<<blocks, THREADS, LDS_BYTES, stream>>>(
      x, W1, b1, cbk, qz, h, payload, aux, loss);
}